// NCN_30734785970432
// MI455X (gfx1250) — compile-verified
//
#include <hip/hip_runtime.h>

// ---------------------------------------------------------------------------
// CBM pipeline for MI455X (gfx1250, wave32).
// Factorized GAT: node = probs * concept_bank  =>  xw = probs * (concept_bank @ gat_w^T)
// so the batched 33.6-GFLOP GEMM collapses to one 0.52-GFLOP batch-independent
// GEMM (cbw, 4.1MB, L2-resident) + 12 FMAs/element. out[64,1000,256] is never
// materialized (recomputed in two passes around the concept softmax).
// GEMMs: fp32 V_WMMA_F32_16X16X4_F32, one wave per 16x16 tile.
// GAT passes: 64KB LDS ring-window of cbw rows staged with
// global_load_async_to_lds_b128 (ASYNCcnt) -> 3x reuse, ds_load inner loop.
// ---------------------------------------------------------------------------

typedef __attribute__((ext_vector_type(2))) float v2f;
typedef __attribute__((ext_vector_type(8))) float v8f;

#define NCN_B    64
#define NCN_NC   1000
#define NCN_C    256
#define NCN_H    4
#define NCN_HC   1024      // H*C
#define NCN_BNI  0.9999950000374996f   // 1/sqrt(1+1e-5)
#define NCN_TILE 14        // n's per GAT block
#define NCN_WIN  16        // LDS window rows = TILE + 2 halo
#define NCN_NCHK 72        // ceil(NC / TILE)

enum { ACT_NONE = 0, ACT_TANH = 1, ACT_RELU = 2 };

__device__ __forceinline__ float lrelu02(float x) { return x > 0.f ? x : 0.2f * x; }

// Async DMA: one b128 element from global into LDS (ASYNCcnt-tracked).
__device__ __forceinline__ void async_copy_b128(unsigned lds_byte_off, const void* gptr)
{
    asm volatile("global_load_async_to_lds_b128 %0, %1, off"
                 :: "v"(lds_byte_off), "v"(gptr)
                 : "memory");
}
__device__ __forceinline__ void wait_async0()
{
    asm volatile("s_wait_asynccnt 0x0" ::: "memory");
}

// ---------------------------------------------------------------------------
// D[m,n] = act( sum_k A[m,k]*W[n,k] + bias[n] )   (A @ W^T, both row-major)
// grid = (ceil(Ma/16), ceil(Nw/16)), block = 32 (one wave -> EXEC all ones).
// ISA 7.12.2 fp32 fragments: lane L holds row (L&15); K-pair at kk + 2*(L>>4).
// ---------------------------------------------------------------------------
template <int ACT>
__global__ __launch_bounds__(32) void wmma_gemm_nt(
    const float* __restrict__ A, const float* __restrict__ W,
    const float* __restrict__ bias, float* __restrict__ D,
    int Ma, int Nw, int K)
{
    const int lane = threadIdx.x & 31;
    const int hi   = lane >> 4;       // selects K-pair {0,1} vs {2,3}
    const int r16  = lane & 15;

    int m = blockIdx.x * 16 + r16; if (m >= Ma) m = Ma - 1;   // clamp: garbage lands
    int n = blockIdx.y * 16 + r16; if (n >= Nw) n = Nw - 1;   // only in masked stores
    const float* __restrict__ arow = A + (size_t)m * K;
    const float* __restrict__ wrow = W + (size_t)n * K;

    v8f acc = {0.f, 0.f, 0.f, 0.f, 0.f, 0.f, 0.f, 0.f};

    for (int kk = 0; kk < K; kk += 4) {
        const int ko = kk + (hi << 1);
        v2f a = *reinterpret_cast<const v2f*>(arow + ko);   // global_load_b64
        v2f b = *reinterpret_cast<const v2f*>(wrow + ko);
        // speculative near-cache prefetch (invalid translations silently dropped)
        __builtin_prefetch(arow + kk + 32, 0, 3);
        __builtin_prefetch(wrow + kk + 32, 0, 3);
        // (neg_a, A, neg_b, B, c_mod, C, reuse_a, reuse_b)
        acc = __builtin_amdgcn_wmma_f32_16x16x4_f32(
            false, a, false, b, (short)0, acc, false, false);
    }

    const int col = blockIdx.y * 16 + r16;
    if (col < Nw) {
        const float bv = bias ? bias[col] : 0.f;
        const int row0 = blockIdx.x * 16 + (hi << 3);
        #pragma unroll
        for (int r = 0; r < 8; ++r) {
            const int row = row0 + r;                 // D VGPR r -> M = r + 8*hi
            if (row < Ma) {
                float v = acc[r] + bv;
                if (ACT == ACT_TANH) v = tanhf(v);
                if (ACT == ACT_RELU) v = fmaxf(v, 0.f);
                D[(size_t)row * Nw + col] = v;
            }
        }
    }
}

// ---------------------------------------------------------------------------
// Row softmax over [R, N]; one 256-thread block per row.
// ---------------------------------------------------------------------------
__global__ __launch_bounds__(256) void softmax_rows(
    const float* __restrict__ in, float* __restrict__ out, int N)
{
    __shared__ float red[256];
    const int row = blockIdx.x, tid = threadIdx.x;
    const float* x = in + (size_t)row * N;

    float m = -3.4e38f;
    for (int i = tid; i < N; i += 256) m = fmaxf(m, x[i]);
    red[tid] = m; __syncthreads();
    for (int s = 128; s > 0; s >>= 1) { if (tid < s) red[tid] = fmaxf(red[tid], red[tid + s]); __syncthreads(); }
    m = red[0]; __syncthreads();

    float sum = 0.f;
    for (int i = tid; i < N; i += 256) sum += expf(x[i] - m);
    red[tid] = sum; __syncthreads();
    for (int s = 128; s > 0; s >>= 1) { if (tid < s) red[tid] += red[tid + s]; __syncthreads(); }
    const float inv = 1.f / red[0];

    for (int i = tid; i < N; i += 256) out[(size_t)row * N + i] = expf(x[i] - m) * inv;
}

// ---------------------------------------------------------------------------
// cs[n,h] = cbw[n,h,:]·att_src[h,:], cd likewise. grid=(NC,H), block=256.
// ---------------------------------------------------------------------------
__global__ __launch_bounds__(256) void head_dots(
    const float* __restrict__ cbw, const float* __restrict__ att_src,
    const float* __restrict__ att_dst, float* __restrict__ cs, float* __restrict__ cd)
{
    __shared__ float red[256];
    const int n = blockIdx.x, h = blockIdx.y, tid = threadIdx.x;
    const float v = cbw[(size_t)n * NCN_HC + h * NCN_C + tid];

    red[tid] = v * att_src[h * NCN_C + tid]; __syncthreads();
    for (int s = 128; s > 0; s >>= 1) { if (tid < s) red[tid] += red[tid + s]; __syncthreads(); }
    if (tid == 0) cs[n * NCN_H + h] = red[0];
    __syncthreads();

    red[tid] = v * att_dst[h * NCN_C + tid]; __syncthreads();
    for (int s = 128; s > 0; s >>= 1) { if (tid < s) red[tid] += red[tid + s]; __syncthreads(); }
    if (tid == 0) cd[n * NCN_H + h] = red[0];
}

// Per-(b,n) GAT attention coefficients folded with source probs and 1/H.
// coef[h*3+s] multiplies cbw[src(s), h, :]; src = {n-1, n+1, n} (ring).
__device__ __forceinline__ void gat_coef(
    const float* __restrict__ pb, const float* __restrict__ cs,
    const float* __restrict__ cd, int n, int np, int nn, int h, float* s_coef)
{
    const float pp = pb[np], pc = pb[n], pn = pb[nn];
    const float ad = pc * cd[n * NCN_H + h];
    const float l0 = lrelu02(pp * cs[np * NCN_H + h] + ad);
    const float l1 = lrelu02(pn * cs[nn * NCN_H + h] + ad);
    const float l2 = lrelu02(pc * cs[n  * NCN_H + h] + ad);
    const float mx = fmaxf(l0, fmaxf(l1, l2));
    const float x0 = expf(l0 - mx), x1 = expf(l1 - mx), x2 = expf(l2 - mx);
    const float inv = 0.25f / (x0 + x1 + x2);   // 0.25 = head mean
    s_coef[h * 3 + 0] = x0 * inv * pp;
    s_coef[h * 3 + 1] = x1 * inv * pn;
    s_coef[h * 3 + 2] = x2 * inv * pc;
}

// Stage cbw rows [n0-1, n0+TILE] (ring-wrapped) into the 64KB LDS window.
__device__ __forceinline__ void stage_window(
    float* sbuf, const float* __restrict__ cbw, int n0, int tid)
{
    const unsigned lds0 = (unsigned)(uintptr_t)&sbuf[0];   // low 32b = LDS offset
    #pragma unroll
    for (int i = 0; i < NCN_WIN; ++i) {
        const int q  = tid + (i << 8);                     // float4 index in window
        const int w  = q >> 8;                             // window row
        const int f4 = q & 255;                            // float4 within row
        const int row = (n0 - 1 + w + NCN_NC) % NCN_NC;    // ring wrap at load time
        async_copy_b128(lds0 + (unsigned)(q << 4),
                        (const void*)(cbw + ((size_t)row << 10) + ((size_t)f4 << 2)));
    }
    wait_async0();
}

// out element from the LDS window: rows (n-n0)=prev, +1=self, +2=next.
__device__ __forceinline__ float gat_out_elem_lds(
    const float* sbuf, const float* s_coef, int wrel, int c,
    float gbias, float g, float be)
{
    const float* w0 = sbuf + (wrel << 10);
    float o = gbias;
    #pragma unroll
    for (int h = 0; h < NCN_H; ++h) {
        o += s_coef[h * 3 + 0] * w0[h * NCN_C + c]                 // prev
           + s_coef[h * 3 + 1] * w0[2 * NCN_HC + h * NCN_C + c]    // next
           + s_coef[h * 3 + 2] * w0[NCN_HC + h * NCN_C + c];       // self
    }
    return fmaxf(o * g + be, 0.f);   // BN(eval) + ReLU
}

// ---------------------------------------------------------------------------
// Pass A: aggregation logits t[b,n] = out[b,n,:]·agg_w + agg_b (out on the fly).
// grid = (B, NCHK), block = 256 (one thread per channel c).
// ---------------------------------------------------------------------------
__global__ __launch_bounds__(256) void gat_logits(
    const float* __restrict__ probs, const float* __restrict__ cbw,
    const float* __restrict__ cs, const float* __restrict__ cd,
    const float* __restrict__ gat_bias, const float* __restrict__ bn_g,
    const float* __restrict__ bn_b, const float* __restrict__ agg_w,
    const float* __restrict__ agg_b, float* __restrict__ tlog)
{
    __shared__ float sbuf[NCN_WIN * NCN_HC];   // 64KB rolling window of cbw rows
    __shared__ float s_coef[12];
    __shared__ float red[256];
    const int b = blockIdx.x, tid = threadIdx.x;
    const int n0 = blockIdx.y * NCN_TILE;
    const float* pb = probs + (size_t)b * NCN_NC;
    const float gbias = gat_bias[tid], g = bn_g[tid] * NCN_BNI, be = bn_b[tid];
    const float aw = agg_w[tid], ab = agg_b[0];

    stage_window(sbuf, cbw, n0, tid);
    __syncthreads();

    const int nend = (n0 + NCN_TILE < NCN_NC) ? n0 + NCN_TILE : NCN_NC;
    for (int n = n0; n < nend; ++n) {
        const int np = (n == 0) ? NCN_NC - 1 : n - 1;
        const int nn = (n == NCN_NC - 1) ? 0 : n + 1;
        if (tid < NCN_H) gat_coef(pb, cs, cd, n, np, nn, tid, s_coef);
        __syncthreads();
        const float v = gat_out_elem_lds(sbuf, s_coef, n - n0, tid, gbias, g, be);
        red[tid] = v * aw; __syncthreads();
        for (int s = 128; s > 0; s >>= 1) { if (tid < s) red[tid] += red[tid + s]; __syncthreads(); }
        if (tid == 0) tlog[(size_t)b * NCN_NC + n] = red[0] + ab;
        __syncthreads();
    }
}

// ---------------------------------------------------------------------------
// Pass B: refined[b,:] += sum_n attw[b,n] * out[b,n,:] (out recomputed).
// grid = (B, NCHK), block = 256; one atomicAdd per (block, channel).
// ---------------------------------------------------------------------------
__global__ __launch_bounds__(256) void gat_aggregate(
    const float* __restrict__ probs, const float* __restrict__ cbw,
    const float* __restrict__ cs, const float* __restrict__ cd,
    const float* __restrict__ gat_bias, const float* __restrict__ bn_g,
    const float* __restrict__ bn_b, const float* __restrict__ attw,
    float* __restrict__ refined)
{
    __shared__ float sbuf[NCN_WIN * NCN_HC];
    __shared__ float s_coef[12];
    const int b = blockIdx.x, tid = threadIdx.x;
    const int n0 = blockIdx.y * NCN_TILE;
    const float* pb = probs + (size_t)b * NCN_NC;
    const float gbias = gat_bias[tid], g = bn_g[tid] * NCN_BNI, be = bn_b[tid];

    stage_window(sbuf, cbw, n0, tid);
    __syncthreads();

    float acc = 0.f;
    const int nend = (n0 + NCN_TILE < NCN_NC) ? n0 + NCN_TILE : NCN_NC;
    for (int n = n0; n < nend; ++n) {
        const int np = (n == 0) ? NCN_NC - 1 : n - 1;
        const int nn = (n == NCN_NC - 1) ? 0 : n + 1;
        if (tid < NCN_H) gat_coef(pb, cs, cd, n, np, nn, tid, s_coef);
        __syncthreads();
        const float v = gat_out_elem_lds(sbuf, s_coef, n - n0, tid, gbias, g, be);
        acc += attw[(size_t)b * NCN_NC + n] * v;
        __syncthreads();
    }
    atomicAdd(&refined[(size_t)b * NCN_C + tid], acc);
}

__global__ void zero_f32(float* __restrict__ p, int n)
{
    const int i = blockIdx.x * blockDim.x + threadIdx.x;
    if (i < n) p[i] = 0.f;
}

// ---------------------------------------------------------------------------
extern "C" void kernel_launch(void* const* d_in, const int* in_sizes, int n_in,
                              void* d_out, int out_size, void* d_ws, size_t ws_size,
                              hipStream_t stream)
{
    const float* features     = (const float*)d_in[0];
    const float* concept_bank = (const float*)d_in[1];
    const float* W1  = (const float*)d_in[2];
    const float* b1  = (const float*)d_in[3];
    const float* W2  = (const float*)d_in[4];
    const float* b2  = (const float*)d_in[5];
    const float* gat_w    = (const float*)d_in[6];
    const float* att_src  = (const float*)d_in[7];
    const float* att_dst  = (const float*)d_in[8];
    const float* gat_bias = (const float*)d_in[9];
    const float* bn_gamma = (const float*)d_in[10];
    const float* bn_beta  = (const float*)d_in[11];
    const float* agg_w    = (const float*)d_in[12];
    const float* agg_b    = (const float*)d_in[13];
    const float* Wc1 = (const float*)d_in[14];
    const float* bc1 = (const float*)d_in[15];
    const float* Wc2 = (const float*)d_in[16];
    const float* bc2 = (const float*)d_in[17];

    // workspace layout (fp32, all offsets 128B-aligned) ~5.4 MB total
    float* ws      = (float*)d_ws;
    float* hidden  = ws;                    // 64*256
    float* scores  = hidden  + 64 * 256;    // 64*1000
    float* probs   = scores  + 64 * 1000;
    float* cbw     = probs   + 64 * 1000;   // 1000*1024
    float* cs      = cbw     + 1000 * 1024; // 1000*4
    float* cd      = cs      + 1000 * 4;
    float* tlog    = cd      + 1000 * 4;    // 64*1000
    float* attw    = tlog    + 64 * 1000;
    float* refined = attw    + 64 * 1000;   // 64*256
    float* h2      = refined + 64 * 256;    // 64*512
    float* logits  = (float*)d_out;         // 64*1000

    // 1) hidden = tanh(features @ W1^T + b1)          [64,256], K=768
    wmma_gemm_nt<ACT_TANH><<<dim3(4, 16), 32, 0, stream>>>(features, W1, b1, hidden, 64, 256, 768);
    // 2) scores = hidden @ W2^T + b2                  [64,1000], K=256
    wmma_gemm_nt<ACT_NONE><<<dim3(4, 63), 32, 0, stream>>>(hidden, W2, b2, scores, 64, 1000, 256);
    // 3) probs = softmax(scores, axis=1)
    softmax_rows<<<64, 256, 0, stream>>>(scores, probs, NCN_NC);
    // 4) cbw = concept_bank @ gat_w^T (batch-independent!)  [1000,1024], K=256
    wmma_gemm_nt<ACT_NONE><<<dim3(63, 64), 32, 0, stream>>>(concept_bank, gat_w, nullptr, cbw, 1000, 1024, 256);
    // 5) cs/cd head projections
    head_dots<<<dim3(NCN_NC, NCN_H), 256, 0, stream>>>(cbw, att_src, att_dst, cs, cd);
    // 6) pass A: aggregation logits (LDS-windowed)
    gat_logits<<<dim3(NCN_B, NCN_NCHK), 256, 0, stream>>>(probs, cbw, cs, cd, gat_bias,
                                                          bn_gamma, bn_beta, agg_w, agg_b, tlog);
    // 7) attw = softmax over concepts
    softmax_rows<<<64, 256, 0, stream>>>(tlog, attw, NCN_NC);
    // 8) refined = sum_n attw * out   (zero + atomic accumulate)
    zero_f32<<<64, 256, 0, stream>>>(refined, 64 * 256);
    gat_aggregate<<<dim3(NCN_B, NCN_NCHK), 256, 0, stream>>>(probs, cbw, cs, cd, gat_bias,
                                                             bn_gamma, bn_beta, attw, refined);
    // 9) h2 = relu(refined @ Wc1^T + bc1)             [64,512], K=256
    wmma_gemm_nt<ACT_RELU><<<dim3(4, 32), 32, 0, stream>>>(refined, Wc1, bc1, h2, 64, 512, 256);
    // 10) logits = h2 @ Wc2^T + bc2                   [64,1000], K=512
    wmma_gemm_nt<ACT_NONE><<<dim3(4, 63), 32, 0, stream>>>(h2, Wc2, bc2, logits, 64, 1000, 512);

    (void)in_sizes; (void)n_in; (void)out_size; (void)ws_size;
}